// HistogramLoss_43516608643710
// MI455X (gfx1250) — compile-verified
//
#include <hip/hip_runtime.h>
#include <hip/hip_bf16.h>

#define RS_BLOCK 256
#define RS_WAVES (RS_BLOCK / 32)
#define RS_ITEMS 16
#define RS_TILE  (RS_BLOCK * RS_ITEMS)   // 4096 elements per tile

typedef unsigned __attribute__((ext_vector_type(4))) v4u;
typedef int      __attribute__((ext_vector_type(4))) v4i;
typedef int      __attribute__((ext_vector_type(8))) v8i;
typedef float    __attribute__((ext_vector_type(2))) v2f;
typedef float    __attribute__((ext_vector_type(8))) v8f;

// ---- float <-> order-preserving uint32 key ----
__device__ __forceinline__ unsigned f2k(float f) {
    unsigned u = __float_as_uint(f);
    unsigned m = (u & 0x80000000u) ? 0xFFFFFFFFu : 0x80000000u;
    return u ^ m;
}
__device__ __forceinline__ float k2f(unsigned k) {
    unsigned m = (k & 0x80000000u) ? 0x80000000u : 0xFFFFFFFFu;
    return __uint_as_float(k ^ m);
}

// ---- TDM: DMA a contiguous run of uint32 from global into LDS.
//      1-D descriptor: tile_dim0 = count elements, tensor_dim0 = valid
//      (OOB tail zero-fills). Issued by one wave; completion via TENSORcnt. ----
__device__ __forceinline__ void tdm_load_tile_u32(const unsigned* gsrc,
                                                  unsigned* lds_dst,
                                                  unsigned count, unsigned valid) {
    unsigned long long ga = (unsigned long long)(uintptr_t)gsrc;
    unsigned lds_off = (unsigned)(uintptr_t)lds_dst;   // low 32 bits = LDS byte offset

    v4u g0;
    g0.x = 1u;                                          // count=1 (valid user descriptor)
    g0.y = lds_off;                                     // lds_addr
    g0.z = (unsigned)(ga & 0xFFFFFFFFu);                // global_addr[31:0]
    g0.w = (unsigned)((ga >> 32) & 0x01FFFFFFu) | (2u << 30);  // addr[56:32] | type=2

    v8i g1;
    g1[0] = (int)(2u << 16);                            // data_size = 2 (4 bytes)
    g1[1] = (int)((valid & 0xFFFFu) << 16);             // tensor_dim0[15:0]
    g1[2] = (int)((valid >> 16) & 0xFFFFu);             // tensor_dim0[31:16], tensor_dim1=0
    g1[3] = (int)((count & 0xFFFFu) << 16);             // tile_dim0
    g1[4] = 0;                                          // tile_dim1=0, tile_dim2=0 (unused)
    g1[5] = 0;                                          // tensor_dim0_stride (unused, 1-D)
    g1[6] = 0;
    g1[7] = 0;

    v4i g2 = {0, 0, 0, 0};
    v4i g3 = {0, 0, 0, 0};
#if defined(__clang_major__) && (__clang_major__ >= 23)
    v8i g4 = {0, 0, 0, 0, 0, 0, 0, 0};
    __builtin_amdgcn_tensor_load_to_lds(g0, g1, g2, g3, g4, 0);
#else
    __builtin_amdgcn_tensor_load_to_lds(g0, g1, g2, g3, 0);
#endif
}

// ---- transforms ----
__global__ void rs_transform_kv(const float* __restrict__ in,
                                unsigned* __restrict__ keys,
                                unsigned* __restrict__ vals, int n) {
    int i = blockIdx.x * blockDim.x + threadIdx.x;
    if (i < n) { keys[i] = f2k(in[i]); vals[i] = (unsigned)i; }
}
__global__ void rs_transform_k(const float* __restrict__ in,
                               unsigned* __restrict__ keys, int n) {
    int i = blockIdx.x * blockDim.x + threadIdx.x;
    if (i < n) keys[i] = f2k(in[i]);
}

// ---- per-block digit histogram, digit-major layout hist[d*nblocks+b] ----
__global__ __launch_bounds__(RS_BLOCK) void rs_hist(const unsigned* __restrict__ keys, int n,
                                                    unsigned* __restrict__ hist,
                                                    int nblocks, int shift) {
    __shared__ unsigned sh[256];
    const int t = threadIdx.x;
    sh[t] = 0u;
    __syncthreads();
    const int base = blockIdx.x * RS_TILE;
    for (int j = 0; j < RS_ITEMS; ++j) {
        int g = base + j * RS_BLOCK + t;
        if (g < n) {
            unsigned d = (keys[g] >> shift) & 0xFFu;
            atomicAdd(&sh[d], 1u);
        }
    }
    __syncthreads();
    hist[(size_t)t * nblocks + blockIdx.x] = sh[t];
}

// ---- per-digit totals across blocks ----
__global__ __launch_bounds__(256) void rs_digit_totals(const unsigned* __restrict__ hist,
                                                       int nblocks,
                                                       unsigned* __restrict__ totals) {
    __shared__ unsigned sh[256];
    const int d = blockIdx.x, t = threadIdx.x;
    unsigned s = 0u;
    for (int b = t; b < nblocks; b += 256) s += hist[(size_t)d * nblocks + b];
    sh[t] = s;
    __syncthreads();
    for (int off = 128; off > 0; off >>= 1) {
        if (t < off) sh[t] += sh[t + off];
        __syncthreads();
    }
    if (t == 0) totals[d] = sh[0];
}

// ---- exclusive scan of 256 digit totals ----
__global__ __launch_bounds__(256) void rs_digit_base(const unsigned* __restrict__ totals,
                                                     unsigned* __restrict__ dbase) {
    __shared__ unsigned sh[256];
    const int t = threadIdx.x;
    unsigned v = totals[t];
    sh[t] = v;
    __syncthreads();
    for (int off = 1; off < 256; off <<= 1) {
        unsigned x = (t >= off) ? sh[t - off] : 0u;
        __syncthreads();
        sh[t] += x;
        __syncthreads();
    }
    dbase[t] = sh[t] - v;   // exclusive
}

// ---- per-digit exclusive scan across blocks (in place) + digit base ----
__global__ __launch_bounds__(256) void rs_row_scan(unsigned* __restrict__ hist, int nblocks,
                                                   const unsigned* __restrict__ dbase) {
    __shared__ unsigned sh[256];
    const int d = blockIdx.x, t = threadIdx.x;
    unsigned carry = dbase[d];
    for (int c = 0; c < nblocks; c += 256) {
        int b = c + t;
        if (c + 256 < nblocks)   // prefetch next chunk (global_prefetch_b8)
            __builtin_prefetch(&hist[(size_t)d * nblocks + c + 256 + t], 0, 1);
        unsigned v = (b < nblocks) ? hist[(size_t)d * nblocks + b] : 0u;
        sh[t] = v;
        __syncthreads();
        for (int off = 1; off < 256; off <<= 1) {
            unsigned x = (t >= off) ? sh[t - off] : 0u;
            __syncthreads();
            sh[t] += x;
            __syncthreads();
        }
        unsigned incl = sh[t];
        unsigned tot = sh[255];
        __syncthreads();
        if (b < nblocks) hist[(size_t)d * nblocks + b] = carry + incl - v;
        carry += tot;
    }
}

// ---- stable scatter: TDM tile load -> 8 stable 1-bit splits (wave32 shfl scans)
//      -> segment-coalesced global writes using scanned offsets ----
template <bool KV>
__global__ __launch_bounds__(RS_BLOCK) void rs_scatter(const unsigned* __restrict__ keys_in,
                                                       const unsigned* __restrict__ vals_in,
                                                       unsigned* __restrict__ keys_out,
                                                       unsigned* __restrict__ vals_out,
                                                       const unsigned* __restrict__ hist,
                                                       int nblocks, int shift, int n) {
    __shared__ unsigned lkeys[RS_TILE];
    __shared__ unsigned lvals[RS_TILE];
    __shared__ unsigned wsum[RS_WAVES];
    __shared__ unsigned wbase[RS_WAVES];
    __shared__ unsigned ztot;
    __shared__ unsigned gbase[256];
    __shared__ unsigned dstart[256];
    const int t = threadIdx.x;
    const int lane = t & 31;
    const int wid = t >> 5;
    const int base = blockIdx.x * RS_TILE;
    int valid = n - base;
    if (valid > RS_TILE) valid = RS_TILE;

    // async DMA the tile(s) into LDS (one wave issues; EXEC-uniform branch)
    if (t < 32) {
        tdm_load_tile_u32(keys_in + base, lkeys, RS_TILE, (unsigned)valid);
        if (KV) tdm_load_tile_u32(vals_in + base, lvals, RS_TILE, (unsigned)valid);
        __builtin_amdgcn_s_wait_tensorcnt(0);
    }
    __syncthreads();
    // pad tail with max keys (stably sorts to the end; vals tail zero-filled by TDM)
    for (int j = 0; j < RS_ITEMS; ++j) {
        int idx = j * RS_BLOCK + t;
        if (idx >= valid) lkeys[idx] = 0xFFFFFFFFu;
    }

    // 8 stable 1-bit splits over the current digit
    for (int b = 0; b < 8; ++b) {
        const unsigned bmask = 1u << (shift + b);
        __syncthreads();
        unsigned k[RS_ITEMS], v[RS_ITEMS];
        int zc = 0;
        const int bi = t * RS_ITEMS;
        for (int j = 0; j < RS_ITEMS; ++j) {
            k[j] = lkeys[bi + j];
            if (KV) v[j] = lvals[bi + j];
            zc += ((k[j] & bmask) == 0u) ? 1 : 0;
        }
        // wave32 inclusive scan of per-thread zero counts
        unsigned incl = (unsigned)zc;
        #pragma unroll
        for (int d2 = 1; d2 < 32; d2 <<= 1) {
            unsigned y = __shfl_up(incl, (unsigned)d2, 32);
            if (lane >= d2) incl += y;
        }
        if (lane == 31) wsum[wid] = incl;
        __syncthreads();
        if (wid == 0) {
            unsigned v8 = (lane < RS_WAVES) ? wsum[lane] : 0u;
            unsigned s8 = v8;
            #pragma unroll
            for (int d2 = 1; d2 < RS_WAVES; d2 <<= 1) {
                unsigned y = __shfl_up(s8, (unsigned)d2, 32);
                if (lane >= d2) s8 += y;
            }
            if (lane < RS_WAVES) wbase[lane] = s8 - v8;
            if (lane == RS_WAVES - 1) ztot = s8;
        }
        __syncthreads();
        unsigned Z = ztot;                               // total zeros in tile
        unsigned zrun = wbase[wid] + incl - (unsigned)zc; // zeros strictly before my items
        for (int j = 0; j < RS_ITEMS; ++j) {
            int i = bi + j;
            unsigned dest;
            if ((k[j] & bmask) == 0u) {
                dest = zrun++;
            } else {
                dest = Z + (unsigned)i - zrun;           // ones-before = i - zeros-before
            }
            lkeys[dest] = k[j];
            if (KV) lvals[dest] = v[j];
        }
    }
    __syncthreads();

    // global offsets for my block + local digit segment starts
    gbase[t] = hist[(size_t)t * nblocks + blockIdx.x];
    {
        const int bi = t * RS_ITEMS;
        for (int j = 0; j < RS_ITEMS; ++j) {
            int i = bi + j;
            unsigned d = (lkeys[i] >> shift) & 0xFFu;
            bool boundary = (i == 0) || (((lkeys[i - 1] >> shift) & 0xFFu) != d);
            if (boundary) dstart[d] = (unsigned)i;
        }
    }
    __syncthreads();

    // segment-coalesced stores; padded elements (stably at the tail) are skipped
    for (int j = 0; j < RS_ITEMS; ++j) {
        int i = j * RS_BLOCK + t;
        if (i < valid) {
            unsigned kk = lkeys[i];
            unsigned d = (kk >> shift) & 0xFFu;
            unsigned pos = gbase[d] + (unsigned)i - dstart[d];
            keys_out[pos] = kk;
            if (KV) vals_out[pos] = lvals[i];
        }
    }
}

// ---- final gather + WMMA-based deterministic squared-error reduction ----
__global__ __launch_bounds__(RS_BLOCK) void hm_gather(const unsigned* __restrict__ sortedIdx,
                                                      const unsigned* __restrict__ tmplKeys,
                                                      const float* __restrict__ input,
                                                      float* __restrict__ out,
                                                      float* __restrict__ partials,
                                                      int n, int m) {
    const int t = threadIdx.x;
    const int lane = t & 31;
    const int wid = t >> 5;
    const int base = blockIdx.x * RS_TILE;
    const float invN = 1.0f / (float)n;
    float acc = 0.0f;
    for (int j = 0; j < RS_ITEMS; ++j) {
        int i = base + j * RS_BLOCK + t;
        if (i < n) {
            // rank = i+1 ; q = rank/N ; position on template grid p = q*M - 1
            float q = (float)(i + 1) * invN;
            float p = q * (float)m - 1.0f;
            float mv;
            if (p <= 0.0f) {
                mv = k2f(tmplKeys[0]);
            } else if (p >= (float)(m - 1)) {
                mv = k2f(tmplKeys[m - 1]);
            } else {
                int j0 = (int)p;
                float w = p - (float)j0;
                float a = k2f(tmplKeys[j0]);
                if (w == 0.0f) {
                    mv = a;   // exact grid hit: the N==M case
                } else {
                    float b2 = k2f(tmplKeys[j0 + 1]);
                    mv = a + w * (b2 - a);
                }
            }
            unsigned idx = sortedIdx[i];
            out[idx] = mv;
            float dd = mv - input[idx];
            acc += dd * dd;
        }
    }
    // wave reduction via V_WMMA_F32_16X16X4_F32 with B = ones:
    // A[m][k]: lane L<16 holds {A[L][0]=acc, A[L][1]=0}, lane L>=16 {A[L-16][2]=acc, A[L-16][3]=0}
    // => D[m][n] = acc_m + acc_{m+16} for every column n.
    // Lanes 0..15 hold D rows 0..7 (VGPR j = row j); lanes 16..31 hold rows 8..15.
    v2f a; a.x = acc;  a.y = 0.0f;
    v2f b; b.x = 1.0f; b.y = 1.0f;
    v8f c = {};
    c = __builtin_amdgcn_wmma_f32_16x16x4_f32(false, a, false, b, (short)0, c, false, false);
    float s = c[0] + c[1] + c[2] + c[3] + c[4] + c[5] + c[6] + c[7];
    float waveTot = __shfl(s, 0, 32) + __shfl(s, 16, 32);
    __shared__ float wtot[RS_WAVES];
    if (lane == 0) wtot[wid] = waveTot;
    __syncthreads();
    if (t == 0) {
        float tot = 0.0f;
        for (int w = 0; w < RS_WAVES; ++w) tot += wtot[w];
        partials[blockIdx.x] = tot;
    }
}

__global__ __launch_bounds__(RS_BLOCK) void hm_finalize(const float* __restrict__ partials,
                                                        int nparts,
                                                        float* __restrict__ lossOut, int n) {
    __shared__ float sh[RS_BLOCK];
    const int t = threadIdx.x;
    float s = 0.0f;
    for (int i = t; i < nparts; i += RS_BLOCK) s += partials[i];
    sh[t] = s;
    __syncthreads();
    for (int off = RS_BLOCK / 2; off > 0; off >>= 1) {
        if (t < off) sh[t] += sh[t + off];
        __syncthreads();
    }
    if (t == 0) *lossOut = sh[0] / (float)n;
}

extern "C" void kernel_launch(void* const* d_in, const int* in_sizes, int n_in,
                              void* d_out, int out_size, void* d_ws, size_t ws_size,
                              hipStream_t stream) {
    const float* input  = (const float*)d_in[0];
    const float* target = (const float*)d_in[1];
    const int n = in_sizes[0];
    const int m = in_sizes[1];
    const int nblocks = (n + RS_TILE - 1) / RS_TILE;
    const int mblocks = (m + RS_TILE - 1) / RS_TILE;
    const int maxb = (nblocks > mblocks) ? nblocks : mblocks;

    unsigned* ws = (unsigned*)d_ws;
    size_t off = 0;
    unsigned* srcKA = ws + off; off += (size_t)n;
    unsigned* srcKB = ws + off; off += (size_t)n;
    unsigned* srcVA = ws + off; off += (size_t)n;
    unsigned* srcVB = ws + off; off += (size_t)n;
    unsigned* tgtKA = ws + off; off += (size_t)m;
    unsigned* hist  = ws + off; off += (size_t)256 * maxb;
    unsigned* totals = ws + off; off += 256;
    unsigned* dbase  = ws + off; off += 256;
    float* partials  = (float*)(ws + off); off += (size_t)nblocks;
    unsigned* tgtKB = (unsigned*)d_out;   // reuse matched-output area as second ping-pong buf
    float* out = (float*)d_out;

    dim3 blk(RS_BLOCK);

    rs_transform_kv<<<dim3((n + RS_BLOCK - 1) / RS_BLOCK), blk, 0, stream>>>(input, srcKA, srcVA, n);
    rs_transform_k <<<dim3((m + RS_BLOCK - 1) / RS_BLOCK), blk, 0, stream>>>(target, tgtKA, m);

    // ---- source argsort: key+index, 4x8-bit LSD passes (A->B->A->B->A) ----
    unsigned *ka = srcKA, *kb = srcKB, *va = srcVA, *vb = srcVB;
    for (int pass = 0; pass < 4; ++pass) {
        const int shift = pass * 8;
        rs_hist        <<<dim3(nblocks), blk, 0, stream>>>(ka, n, hist, nblocks, shift);
        rs_digit_totals<<<dim3(256),     blk, 0, stream>>>(hist, nblocks, totals);
        rs_digit_base  <<<dim3(1),       blk, 0, stream>>>(totals, dbase);
        rs_row_scan    <<<dim3(256),     blk, 0, stream>>>(hist, nblocks, dbase);
        rs_scatter<true><<<dim3(nblocks), blk, 0, stream>>>(ka, va, kb, vb, hist, nblocks, shift, n);
        unsigned* tk = ka; ka = kb; kb = tk;
        unsigned* tv = va; va = vb; vb = tv;
    }
    // sorted keys in srcKA, sorted original indices in srcVA

    // ---- target sort: keys only ----
    unsigned *ta = tgtKA, *tb = tgtKB;
    for (int pass = 0; pass < 4; ++pass) {
        const int shift = pass * 8;
        rs_hist        <<<dim3(mblocks), blk, 0, stream>>>(ta, m, hist, mblocks, shift);
        rs_digit_totals<<<dim3(256),     blk, 0, stream>>>(hist, mblocks, totals);
        rs_digit_base  <<<dim3(1),       blk, 0, stream>>>(totals, dbase);
        rs_row_scan    <<<dim3(256),     blk, 0, stream>>>(hist, mblocks, dbase);
        rs_scatter<false><<<dim3(mblocks), blk, 0, stream>>>(ta, nullptr, tb, nullptr, hist, mblocks, shift, m);
        unsigned* tt = ta; ta = tb; tb = tt;
    }
    // sorted target keys in tgtKA (d_out alias no longer needed)

    hm_gather  <<<dim3(nblocks), blk, 0, stream>>>(srcVA, tgtKA, input, out, partials, n, m);
    hm_finalize<<<dim3(1),       blk, 0, stream>>>(partials, nblocks, out + n, n);
}